// MoaAttention_3685081940644
// MI455X (gfx1250) — compile-verified
//
#include <hip/hip_runtime.h>
#include <math.h>

// ---------------------------------------------------------------------------
// MoA (mixture-of-attention-heads) forward for MI455X / gfx1250.
// fp32 throughout (matches reference numerics) using V_WMMA_F32_16X16X4_F32.
// 4-wave workgroups (one per SIMD32 of a WGP) sharing LDS tiles; tile loads
// use GLOBAL_LOAD_ASYNC_TO_LDS_B128 when the toolchain exposes the builtin.
// ---------------------------------------------------------------------------

typedef float v2f __attribute__((ext_vector_type(2)));
typedef float v8f __attribute__((ext_vector_type(8)));
typedef int   v4i __attribute__((ext_vector_type(4)));

__device__ __forceinline__ v8f wmma_f32(v2f a, v2f b, v8f c) {
  // (neg_a, A, neg_b, B, c_mod, C, reuse_a, reuse_b)
  return __builtin_amdgcn_wmma_f32_16x16x4_f32(false, a, false, b, (short)0, c,
                                               false, false);
}

// ---- async global->LDS (gfx1250) with compile-safe fallback ---------------
#if defined(__has_builtin)
#if __has_builtin(__builtin_amdgcn_global_load_async_to_lds_b128) && \
    __has_builtin(__builtin_amdgcn_s_wait_asynccnt)
#define USE_ASYNC_LDS 1
#else
#define USE_ASYNC_LDS 0
#endif
#else
#define USE_ASYNC_LDS 0
#endif

typedef __attribute__((address_space(1))) v4i gbl_v4i;  // global b128 unit
typedef __attribute__((address_space(3))) v4i lds_v4i;  // LDS b128 unit

// move 16 bytes (4 floats) global -> LDS for this lane
__device__ __forceinline__ void tile_ld16(float* lds_dst, const float* gsrc) {
#if USE_ASYNC_LDS
  __builtin_amdgcn_global_load_async_to_lds_b128((gbl_v4i*)gsrc,
                                                 (lds_v4i*)lds_dst, 0, 0);
#else
  float4 v4 = *reinterpret_cast<const float4*>(gsrc);
  lds_dst[0] = v4.x; lds_dst[1] = v4.y; lds_dst[2] = v4.z; lds_dst[3] = v4.w;
#endif
}
// drain this wave's async copies, then block barrier
__device__ __forceinline__ void tile_ld_wait() {
#if USE_ASYNC_LDS
  __builtin_amdgcn_s_wait_asynccnt(0);
#endif
  __syncthreads();
}

constexpr int BB   = 4;          // batch
constexpr int T    = 2048;       // sequence
constexpr int D    = 512;        // model dim
constexpr int H    = 8;          // heads
constexpr int E    = 16;         // experts
constexpr int HD   = 64;         // head dim
constexpr int KTOP = 8;          // topk == H
constexpr int N    = BB * T;     // tokens = 8192

// ---------------------------------------------------------------------------
// 1. Gating: logits = x @ expert_gate, softmax, top-8 selection.
// ---------------------------------------------------------------------------
__global__ void gate_topk_kernel(const float* __restrict__ x,
                                 const float* __restrict__ egate,
                                 float* __restrict__ gates,
                                 int* __restrict__ idx) {
  int n = blockIdx.x * blockDim.x + threadIdx.x;
  if (n >= N) return;
  float acc[E];
#pragma unroll
  for (int e = 0; e < E; ++e) acc[e] = 0.f;
  const float* xr = x + (size_t)n * D;
  for (int d = 0; d < D; ++d) {
    float xv = xr[d];
#pragma unroll
    for (int e = 0; e < E; ++e) acc[e] = fmaf(xv, egate[d * E + e], acc[e]);
  }
  float mx = acc[0];
#pragma unroll
  for (int e = 1; e < E; ++e) mx = fmaxf(mx, acc[e]);
  float s = 0.f;
#pragma unroll
  for (int e = 0; e < E; ++e) { acc[e] = __expf(acc[e] - mx); s += acc[e]; }
  float inv = 1.f / s;
#pragma unroll
  for (int e = 0; e < E; ++e) acc[e] *= inv;
#pragma unroll
  for (int j = 0; j < KTOP; ++j) {
    int best = 0; float bv = -1.f;
#pragma unroll
    for (int e = 0; e < E; ++e)
      if (acc[e] > bv) { bv = acc[e]; best = e; }
    gates[n * KTOP + j] = bv;
    idx[n * KTOP + j] = best;
    acc[best] = -2.f;
  }
}

// ---------------------------------------------------------------------------
// 2. Batched fp32 WMMA GEMM: C[z] = A[z] (MxK) @ B[z] (KxN) (+bias).
//    4 waves / block; block tile 64x64 (each wave a 16-row slice); K blocked
//    by 64 through LDS shared by all 4 waves.
//    Requires: M%64==0, N%64==0, K%64==0, lda/ldb%4==0.
// ---------------------------------------------------------------------------
#define KT 64
__global__ void __launch_bounds__(128)
gemm_wmma_kernel(const float* __restrict__ A, int lda, long sA,
                 const float* __restrict__ B, int ldb, long sB,
                 float* __restrict__ C, int ldc, long sC,
                 int Ktot, const float* __restrict__ bias) {
  __shared__ float As[64][KT + 4];   // 64 rows of A (4 wave slices)
  __shared__ float Bs[KT][64 + 4];
  const int tid = threadIdx.x;       // 0..127
  const int lane = tid & 31;
  const int wv = tid >> 5;           // wave id -> M sub-slice
  const int m0 = blockIdx.y * 64;
  const int n0 = blockIdx.x * 64;
  A += (size_t)blockIdx.z * sA;
  B += (size_t)blockIdx.z * sB;
  C += (size_t)blockIdx.z * sC;
  const int row = lane & 15;         // A: M index within slice / B: N index
  const int hl = lane >> 4;          // half-wave
  const int koff = hl * 2;           // K sub-offset within 4-step
  const int mrow = wv * 16 + row;    // A LDS row for fragments

  v8f zero = {0.f, 0.f, 0.f, 0.f, 0.f, 0.f, 0.f, 0.f};
  v8f acc[4];
#pragma unroll
  for (int s = 0; s < 4; ++s) acc[s] = zero;

  for (int kb = 0; kb < Ktot; kb += KT) {
    // A tile 64xKT and B tile KTx64: 8 x b128 per thread each
#pragma unroll
    for (int i = 0; i < 8; ++i) {
      int q = tid + 128 * i;
      int r = q >> 4, c = (q & 15) * 4;
      tile_ld16(&As[r][c], &A[(size_t)(m0 + r) * lda + kb + c]);
    }
#pragma unroll
    for (int i = 0; i < 8; ++i) {
      int q = tid + 128 * i;
      int r = q >> 4, c = (q & 15) * 4;
      tile_ld16(&Bs[r][c], &B[(size_t)(kb + r) * ldb + n0 + c]);
    }
    if (kb + KT < Ktot)  // gfx1250 global_prefetch of next A tile
      __builtin_prefetch(&A[(size_t)(m0 + wv * 16) * lda + kb + KT], 0, 1);
    tile_ld_wait();
#pragma unroll
    for (int ks = 0; ks < KT / 4; ++ks) {
      int kk = ks * 4 + koff;
      v2f a; a.x = As[mrow][kk]; a.y = As[mrow][kk + 1];
#pragma unroll
      for (int s = 0; s < 4; ++s) {
        v2f b; b.x = Bs[kk][s * 16 + row]; b.y = Bs[kk + 1][s * 16 + row];
        acc[s] = wmma_f32(a, b, acc[s]);
      }
    }
    __syncthreads();
  }
  // store: C-frag element v -> row (wv*16 + v + 8*hl), col (lane&15)
#pragma unroll
  for (int s = 0; s < 4; ++s) {
#pragma unroll
    for (int v = 0; v < 8; ++v) {
      int r = wv * 16 + v + 8 * hl;
      int c = n0 + s * 16 + row;
      float o = acc[s][v];
      if (bias) o += bias[c];
      C[(size_t)(m0 + r) * ldc + c] = o;
    }
  }
}

// ---------------------------------------------------------------------------
// 3. Flash attention: 4 waves / block, each wave owns a 16-query tile;
//    all waves share each streamed 16-key K/V tile in LDS (4x reuse).
//    Q gathered from dense_q by top-k expert index. Full (non-causal) attn.
// ---------------------------------------------------------------------------
__global__ void __launch_bounds__(128)
flash_attn_kernel(const float* __restrict__ dense_q,
                  const int* __restrict__ idx,
                  const float* __restrict__ keyt,
                  const float* __restrict__ value,
                  float* __restrict__ ao) {
  __shared__ float Ql[4][16][HD + 4];
  __shared__ float Kl[16][HD + 4];
  __shared__ float Vl[16][HD + 4];
  __shared__ float Pl[4][16][17];    // per-wave S tile, then P tile
  __shared__ float scaleL[4][16];
  __shared__ float linvL[4][16];
  __shared__ int eL[4][16];

  const int tid = threadIdx.x;
  const int lane = tid & 31;
  const int wv = tid >> 5;
  const int t0 = blockIdx.x * 64 + wv * 16;  // this wave's query tile
  const int h = blockIdx.y;
  const int b = blockIdx.z;
  const int row = lane & 15;
  const int hl = lane >> 4;
  const int koff = hl * 2;

  if (lane < 16) eL[wv][lane] = idx[(size_t)(b * T + t0 + lane) * KTOP + h];
  __syncthreads();
  // gather this wave's Q tile: slot h of token t uses expert eL[wv][t-t0]
#pragma unroll
  for (int i = 0; i < 8; ++i) {
    int q = lane + 32 * i;             // 256 float4 over 32 lanes
    int m = q >> 4, c = (q & 15) * 4;
    int n = b * T + t0 + m;
    tile_ld16(&Ql[wv][m][c], &dense_q[((size_t)n * E + eL[wv][m]) * HD + c]);
  }

  float m_r = -INFINITY, l_r = 0.f;    // live on lanes 0..15 (row = lane)
  v8f zero = {0.f, 0.f, 0.f, 0.f, 0.f, 0.f, 0.f, 0.f};
  v8f acc[4];
#pragma unroll
  for (int s = 0; s < 4; ++s) acc[s] = zero;

  for (int s0 = 0; s0 < T; s0 += 16) {
    __syncthreads();  // all waves done with Kl/Vl/Pl from previous iteration
    // K & V tiles 16x64, loaded by all 128 threads (2 x b128 each)
#pragma unroll
    for (int i = 0; i < 2; ++i) {
      int q = tid + 128 * i;
      int r = q >> 4, c = (q & 15) * 4;
      size_t base = ((size_t)(b * T) + s0 + r) * D + h * HD + c;
      tile_ld16(&Kl[r][c], &keyt[base]);
      tile_ld16(&Vl[r][c], &value[base]);
    }
    tile_ld_wait();
    // S = Q @ K^T  (K-dim = HD = 64 -> 16 wmma steps)
    v8f sf = zero;
#pragma unroll
    for (int ks = 0; ks < 16; ++ks) {
      int kk = ks * 4 + koff;
      v2f a; a.x = Ql[wv][row][kk]; a.y = Ql[wv][row][kk + 1];
      v2f bb; bb.x = Kl[row][kk]; bb.y = Kl[row][kk + 1];  // B[k][n]=K[n][k]
      sf = wmma_f32(a, bb, sf);
    }
#pragma unroll
    for (int v = 0; v < 8; ++v) Pl[wv][v + 8 * hl][row] = sf[v] * 0.125f;
    __syncthreads();
    // online softmax per row (lanes 0..15 of each wave, row = lane)
    if (lane < 16) {
      float mx = m_r;
#pragma unroll
      for (int j = 0; j < 16; ++j) mx = fmaxf(mx, Pl[wv][lane][j]);
      float sum = 0.f;
#pragma unroll
      for (int j = 0; j < 16; ++j) {
        float p = __expf(Pl[wv][lane][j] - mx);
        Pl[wv][lane][j] = p;
        sum += p;
      }
      float sc = __expf(m_r - mx);
      l_r = l_r * sc + sum;
      m_r = mx;
      scaleL[wv][lane] = sc;
    }
    __syncthreads();
#pragma unroll
    for (int s = 0; s < 4; ++s)
#pragma unroll
      for (int v = 0; v < 8; ++v) acc[s][v] *= scaleL[wv][v + 8 * hl];
    // O += P @ V  (K-dim = 16 keys -> 4 wmma steps, 4 HD subtiles)
#pragma unroll
    for (int ks = 0; ks < 4; ++ks) {
      int kk = ks * 4 + koff;
      v2f a; a.x = Pl[wv][row][kk]; a.y = Pl[wv][row][kk + 1];
#pragma unroll
      for (int s = 0; s < 4; ++s) {
        v2f bb; bb.x = Vl[kk][s * 16 + row]; bb.y = Vl[kk + 1][s * 16 + row];
        acc[s] = wmma_f32(a, bb, acc[s]);
      }
    }
  }
  if (lane < 16) linvL[wv][lane] = 1.f / l_r;
  __syncthreads();
  // write ao as [B,T,H,HD] == [N,K,HD]
#pragma unroll
  for (int s = 0; s < 4; ++s)
#pragma unroll
    for (int v = 0; v < 8; ++v) {
      int r = v + 8 * hl;
      int t = t0 + r;
      ao[(((size_t)(b * T + t)) * H + h) * HD + s * 16 + row] =
          acc[s][v] * linvL[wv][r];
    }
}

// ---------------------------------------------------------------------------
// 4. Scatter gate-weighted ao into dense [N][E][HD] (top-k indices distinct,
//    every element written so no pre-zero needed).
// ---------------------------------------------------------------------------
__global__ void moe_scatter_kernel(const float* __restrict__ ao,
                                   const float* __restrict__ gates,
                                   const int* __restrict__ idx,
                                   float* __restrict__ mdense) {
  int tid = blockIdx.x * blockDim.x + threadIdx.x;
  if (tid >= N * HD) return;
  int n = tid >> 6;
  int hd = tid & 63;
  float vals[E];
#pragma unroll
  for (int e = 0; e < E; ++e) vals[e] = 0.f;
#pragma unroll
  for (int j = 0; j < KTOP; ++j) {
    int e = idx[n * KTOP + j];
    vals[e] += gates[n * KTOP + j] * ao[((size_t)n * KTOP + j) * HD + hd];
  }
#pragma unroll
  for (int e = 0; e < E; ++e)
    mdense[((size_t)n * E + e) * HD + hd] = vals[e];
}

// ---------------------------------------------------------------------------
extern "C" void kernel_launch(void* const* d_in, const int* in_sizes, int n_in,
                              void* d_out, int out_size, void* d_ws,
                              size_t ws_size, hipStream_t stream) {
  (void)in_sizes; (void)n_in; (void)out_size; (void)ws_size;
  const float* query = (const float*)d_in[0];
  const float* keyt  = (const float*)d_in[1];
  const float* value = (const float*)d_in[2];
  const float* egate = (const float*)d_in[3];
  const float* w1    = (const float*)d_in[4];  // [E, D, HD]
  const float* w2    = (const float*)d_in[5];  // [E, HD, D] == flat [E*HD, D]
  const float* outw  = (const float*)d_in[6];  // [D, D] (in, out)
  const float* outb  = (const float*)d_in[7];  // [D]
  float* out = (float*)d_out;

  char* ws = (char*)d_ws;
  float* gates   = (float*)ws;  ws += (size_t)N * KTOP * sizeof(float);
  int*   idx     = (int*)ws;    ws += (size_t)N * KTOP * sizeof(int);
  float* dense_q = (float*)ws;  ws += (size_t)N * E * HD * sizeof(float);
  float* ao      = (float*)ws;  ws += (size_t)N * H * HD * sizeof(float);
  float* y       = (float*)ws;  ws += (size_t)N * D * sizeof(float);
  float* mdense  = dense_q;  // dense_q dead after attention; reuse (~32MB)

  // 1. gating + top-k
  gate_topk_kernel<<<N / 256, 256, 0, stream>>>(query, egate, gates, idx);

  // 2. dense expert Q-projection: for each expert e, x @ w1[e] -> dense_q[:,e,:]
  gemm_wmma_kernel<<<dim3(HD / 64, N / 64, E), 128, 0, stream>>>(
      query, D, 0L, w1, HD, (long)D * HD, dense_q, E * HD, (long)HD, D,
      nullptr);

  // 3. attention (Q gathered by expert index inside kernel)
  flash_attn_kernel<<<dim3(T / 64, H, BB), 128, 0, stream>>>(dense_q, idx,
                                                             keyt, value, ao);

  // 4. gate-weighted scatter to [N, E*HD]
  moe_scatter_kernel<<<(N * HD) / 256, 256, 0, stream>>>(ao, gates, idx,
                                                         mdense);

  // 5. MoE reduce as one GEMM: y = mdense[N x E*HD] @ w2_flat[E*HD x D]
  gemm_wmma_kernel<<<dim3(D / 64, N / 64, 1), 128, 0, stream>>>(
      mdense, E * HD, 0L, w2, D, 0L, y, D, 0L, E * HD, nullptr);

  // 6. output projection: out = y @ out_w + out_b
  gemm_wmma_kernel<<<dim3(D / 64, N / 64, 1), 128, 0, stream>>>(
      y, D, 0L, outw, D, 0L, out, D, 0L, D, outb);
}